// DilatedSliddingWindowAttention_15839839388320
// MI455X (gfx1250) — compile-verified
//
#include <hip/hip_runtime.h>
#include <hip/hip_bf16.h>

// Problem constants (from reference)
constexpr int B   = 2;
constexpr int S   = 4096;
constexpr int E   = 1024;
constexpr int D   = 64;     // QD == VD == 64
constexpr int WIN = 5;
constexpr int DIL = 2;

typedef __attribute__((ext_vector_type(2))) float v2f;
typedef __attribute__((ext_vector_type(8))) float v8f;

// ---------------------------------------------------------------------------
// Kernel 1: fused projection  out[BS x 64] = x[BS x 1024] @ W[1024 x 64] + b
// One wave32 per 16-row tile; 4 WMMA n-tiles (N=64); K stepped by 4 using
// V_WMMA_F32_16X16X4_F32 (fp32 end-to-end).
//
// A (16x4 f32) per ISA 7.12.2: lane L holds row M=L&15; VGPR0=K(kh), VGPR1=K(kh+1)
// with kh = (L>=16)?2:0.  B (4x16) mirrored: lane L holds col N=L&15, same kh.
// C/D (16x16 f32): VGPR r, lanes 0-15 -> M=r, lanes 16-31 -> M=r+8, N=L&15.
// ---------------------------------------------------------------------------
__global__ __launch_bounds__(256) void qkv_proj_kernel(
    const float* __restrict__ x,      // (B*S, E)
    const float* __restrict__ W,      // (E, D)
    const float* __restrict__ bias,   // (D)
    float* __restrict__ out)          // (B*S, D)
{
    const int lane  = threadIdx.x & 31;
    const int wave  = threadIdx.x >> 5;
    const int mtile = blockIdx.x * 8 + wave;        // 512 tiles total
    const int row0  = mtile * 16;

    const int m  = lane & 15;
    const int n  = lane & 15;
    const int kh = (lane >> 4) * 2;                 // 0 or 2

    const float* xrow = x + (size_t)(row0 + m) * E + kh;

    v8f acc0 = {}, acc1 = {}, acc2 = {}, acc3 = {};

    for (int k0 = 0; k0 < E; k0 += 4) {
        // A fragment: two consecutive K values for this lane's row
        v2f a = *(const v2f*)(xrow + k0);

        const float* wp = W + (size_t)(k0 + kh) * D + n;
        v2f b0, b1, b2, b3;
        b0.x = wp[0];      b0.y = wp[D];            // n-tile 0
        b1.x = wp[16];     b1.y = wp[D + 16];       // n-tile 1
        b2.x = wp[32];     b2.y = wp[D + 32];       // n-tile 2
        b3.x = wp[48];     b3.y = wp[D + 48];       // n-tile 3

        acc0 = __builtin_amdgcn_wmma_f32_16x16x4_f32(false, a, false, b0,
                                                     (short)0, acc0, false, false);
        acc1 = __builtin_amdgcn_wmma_f32_16x16x4_f32(false, a, false, b1,
                                                     (short)0, acc1, false, false);
        acc2 = __builtin_amdgcn_wmma_f32_16x16x4_f32(false, a, false, b2,
                                                     (short)0, acc2, false, false);
        acc3 = __builtin_amdgcn_wmma_f32_16x16x4_f32(false, a, false, b3,
                                                     (short)0, acc3, false, false);
    }

    // Write out with bias. M = r + 8*(lane>=16), N = (lane&15) + 16*t.
    const int mo = (lane >> 4) * 8;
    float* orow = out + (size_t)(row0 + mo) * D + n;
#pragma unroll
    for (int r = 0; r < 8; ++r) {
        float* o = orow + (size_t)r * D;
        o[0]  = acc0[r] + bias[n];
        o[16] = acc1[r] + bias[n + 16];
        o[32] = acc2[r] + bias[n + 32];
        o[48] = acc3[r] + bias[n + 48];
    }
}

// ---------------------------------------------------------------------------
// Kernel 2: Vsum[b][d] = sum_s V[b,s,d]
// ---------------------------------------------------------------------------
__global__ __launch_bounds__(64) void vsum_kernel(
    const float* __restrict__ V, float* __restrict__ Vsum)
{
    const int b = blockIdx.x;
    const int d = threadIdx.x;
    const float* p = V + (size_t)b * S * D + d;
    float s = 0.0f;
#pragma unroll 4
    for (int i = 0; i < S; ++i) s += p[(size_t)i * D];
    Vsum[b * D + d] = s;
}

// ---------------------------------------------------------------------------
// Kernel 3: windowed attention with closed-form softmax over the length-S axis
// (untouched logits are 0, so exp=1):
//   out = (Vsum + sum_{j in win} (e^{q.k_j} - 1) V_j) / (S + sum (e^{q.k_j}-1))
// One wave32 per (b,i); lane owns dims {2*lane, 2*lane+1}.
// ---------------------------------------------------------------------------
__global__ __launch_bounds__(256) void attn_kernel(
    const float* __restrict__ Q, const float* __restrict__ K,
    const float* __restrict__ V, const float* __restrict__ Vsum,
    float* __restrict__ out)
{
    const int gwave = (blockIdx.x * blockDim.x + threadIdx.x) >> 5;
    if (gwave >= B * S) return;
    const int lane = threadIdx.x & 31;
    const int b = gwave / S;
    const int i = gwave - b * S;

    const size_t rowQ = (size_t)(b * S + i) * D;
    const float2 qv = *(const float2*)(Q + rowQ + lane * 2);

    float accx = Vsum[b * D + lane * 2];
    float accy = Vsum[b * D + lane * 2 + 1];
    float Z = (float)S;

#pragma unroll
    for (int w = 0; w < WIN; ++w) {
        const int j = i + DIL * (w - WIN / 2);
        if (j < 0 || j >= S) continue;
        const size_t rowJ = (size_t)(b * S + j) * D;
        const float2 kv = *(const float2*)(K + rowJ + lane * 2);
        float c = qv.x * kv.x + qv.y * kv.y;
        // wave32 butterfly reduction -> all lanes hold the full dot product
#pragma unroll
        for (int off = 16; off > 0; off >>= 1) c += __shfl_xor(c, off, 32);
        const float em1 = __expf(c) - 1.0f;
        const float2 vv = *(const float2*)(V + rowJ + lane * 2);
        accx += em1 * vv.x;
        accy += em1 * vv.y;
        Z    += em1;
    }

    const float inv = 1.0f / Z;
    float2 o;
    o.x = accx * inv;
    o.y = accy * inv;
    *(float2*)(out + rowQ + lane * 2) = o;
}

// ---------------------------------------------------------------------------
// Launch
// ---------------------------------------------------------------------------
extern "C" void kernel_launch(void* const* d_in, const int* in_sizes, int n_in,
                              void* d_out, int out_size, void* d_ws, size_t ws_size,
                              hipStream_t stream) {
    const float* x  = (const float*)d_in[0];
    const float* Wq = (const float*)d_in[1];
    const float* bq = (const float*)d_in[2];
    const float* Wk = (const float*)d_in[3];
    const float* bk = (const float*)d_in[4];
    const float* Wv = (const float*)d_in[5];
    const float* bv = (const float*)d_in[6];
    float* outp = (float*)d_out;

    // Workspace: Q | K | V | Vsum  (floats)
    constexpr size_t QKV = (size_t)B * S * D;      // 524288 floats each
    float* ws = (float*)d_ws;
    float* Qb = ws;
    float* Kb = ws + QKV;
    float* Vb = ws + 2 * QKV;
    float* Vs = ws + 3 * QKV;

    const int mtiles = (B * S) / 16;               // 512
    dim3 gproj(mtiles / 8);                        // 8 waves per 256-thread block
    qkv_proj_kernel<<<gproj, 256, 0, stream>>>(x, Wq, bq, Qb);
    qkv_proj_kernel<<<gproj, 256, 0, stream>>>(x, Wk, bk, Kb);
    qkv_proj_kernel<<<gproj, 256, 0, stream>>>(x, Wv, bv, Vb);

    vsum_kernel<<<B, 64, 0, stream>>>(Vb, Vs);

    const int nwaves = B * S;                      // 8192 waves
    attn_kernel<<<(nwaves * 32) / 256, 256, 0, stream>>>(Qb, Kb, Vb, Vs, outp);
}